// RayObsGraph_45019847196998
// MI455X (gfx1250) — compile-verified
//
#include <hip/hip_runtime.h>
#include <hip/hip_bf16.h>

typedef __attribute__((ext_vector_type(2))) float v2f;
typedef __attribute__((ext_vector_type(8))) float v8f;

#define NB   4        // batch
#define NN   8192     // nodes per graph
#define NE   65536    // edges per graph
#define FOBS 128
#define FH   256
#define FOUT 1024
#define NLOG 18

// ---------------------------------------------------------------------------
// Degree / normalization
// ---------------------------------------------------------------------------
__global__ void k_deg_init(float* __restrict__ deg, int n) {
    int i = blockIdx.x * blockDim.x + threadIdx.x;
    if (i < n) deg[i] = 1.0f;                      // self-loop contributes 1
}

__global__ void k_deg_accum(const long long* __restrict__ edges,
                            float* __restrict__ deg) {
    int t = blockIdx.x * blockDim.x + threadIdx.x;
    int b = t / NE;
    if (b >= NB) return;
    int e = t - b * NE;
    long long d = edges[(size_t)b * 2 * NE + NE + e];   // edges[b,1,e]
    atomicAdd(&deg[b * NN + (int)d], 1.0f);
}

__global__ void k_dinv(const float* __restrict__ deg,
                       float* __restrict__ dinv, int n) {
    int i = blockIdx.x * blockDim.x + threadIdx.x;
    if (i < n) dinv[i] = rsqrtf(deg[i]);           // deg >= 1 always
}

// ---------------------------------------------------------------------------
// Dense GEMM via V_WMMA_F32_16X16X4_F32.
// C[b] = A[b] @ W   (A: M x K per batch, W: K x N shared, C: M x N per batch)
// Block = 128 threads (4 waves); each wave owns one 16-col tile -> 64 cols/blk.
// ---------------------------------------------------------------------------
template <int K, int N>
__global__ void gemm_wmma_f32(const float* __restrict__ A,
                              const float* __restrict__ W,
                              float* __restrict__ C, int M) {
    const int lane  = threadIdx.x & 31;
    const int wave  = threadIdx.x >> 5;            // 0..3
    const int n0    = blockIdx.x * 64 + wave * 16; // col tile base
    const int m0    = blockIdx.y * 16;             // row tile base
    const int b     = blockIdx.z;
    const int mlane = lane & 15;                   // M for A-frag, N for B/C
    const int hi    = lane >> 4;                   // lane half selects K pair

    const float* Ab = A + (size_t)b * M * K;
    float*       Cb = C + (size_t)b * M * N;
    const float* arow = Ab + (size_t)(m0 + mlane) * K;  // A[m, :]
    const float* wcol = W + n0 + mlane;                 // W[:, n]

    v8f acc = {};
#pragma unroll 8
    for (int k = 0; k < K; k += 4) {
        const int ka = k + 2 * hi;
        v2f a, bb;
        a.x  = arow[ka];
        a.y  = arow[ka + 1];
        bb.x = wcol[(size_t)ka * N];
        bb.y = wcol[(size_t)(ka + 1) * N];
        acc = __builtin_amdgcn_wmma_f32_16x16x4_f32(
            false, a, false, bb, (short)0, acc, false, false);
    }
    // C layout: VGPR r -> row m0 + r + 8*hi, col n0 + mlane
    float* cptr = Cb + (size_t)(m0 + 8 * hi) * N + n0 + mlane;
#pragma unroll
    for (int r = 0; r < 8; ++r) cptr[(size_t)r * N] = acc[r];
}

// ---------------------------------------------------------------------------
// agg[b,i,:] = h[b,i,:] * dinv[b,i]^2      (self-loop term, also zero-inits)
// ---------------------------------------------------------------------------
template <int F>
__global__ void k_self_init(const float* __restrict__ h,
                            const float* __restrict__ dinv,
                            float* __restrict__ agg) {
    size_t idx = (size_t)blockIdx.x * blockDim.x + threadIdx.x;
    if (idx >= (size_t)NB * NN * F) return;
    size_t node_g = idx / F;                       // b*NN + node
    float w = dinv[node_g];
    agg[idx] = h[idx] * (w * w);
}

// ---------------------------------------------------------------------------
// Edge scatter: agg[b,dst,:] += h[b,src,:] * dinv[src]*dinv[dst]
// One block per edge; float4 gather, scalar f32 atomics (resolve in L2:
// agg2 = 128MB fits in the 192MB L2).
// ---------------------------------------------------------------------------
template <int F, int TPB>
__global__ void k_scatter(const long long* __restrict__ edges,
                          const float* __restrict__ h,
                          const float* __restrict__ dinv,
                          float* __restrict__ agg) {
    const int e = blockIdx.x;
    const int b = blockIdx.y;
    const long long s = edges[(size_t)b * 2 * NE + e];        // edges[b,0,e]
    const long long d = edges[(size_t)b * 2 * NE + NE + e];   // edges[b,1,e]
    const float w = dinv[b * NN + (int)s] * dinv[b * NN + (int)d];
    const float* hs = h   + ((size_t)b * NN + (size_t)s) * F;
    float*       ad = agg + ((size_t)b * NN + (size_t)d) * F;
#pragma unroll
    for (int f = threadIdx.x * 4; f < F; f += TPB * 4) {
        float4 v = *(const float4*)(hs + f);
        atomicAdd(ad + f + 0, v.x * w);
        atomicAdd(ad + f + 1, v.y * w);
        atomicAdd(ad + f + 2, v.z * w);
        atomicAdd(ad + f + 3, v.w * w);
    }
}

// ---------------------------------------------------------------------------
// In-place bias + ReLU
// ---------------------------------------------------------------------------
template <int F>
__global__ void k_bias_relu(float* __restrict__ x,
                            const float* __restrict__ bias) {
    size_t idx = (size_t)blockIdx.x * blockDim.x + threadIdx.x;
    if (idx >= (size_t)NB * NN * F) return;
    int f = (int)(idx % F);
    x[idx] = fmaxf(x[idx] + bias[f], 0.0f);
}

// ---------------------------------------------------------------------------
// Pack W_logit [1024,18] and W_val [1024,1] into padded W_head [1024,32]
// ---------------------------------------------------------------------------
__global__ void k_prep_head(const float* __restrict__ W_logit,
                            const float* __restrict__ W_val,
                            float* __restrict__ Wh) {
    int idx = blockIdx.x * blockDim.x + threadIdx.x;
    if (idx >= FOUT * 32) return;
    int k = idx >> 5, j = idx & 31;
    float v = 0.0f;
    if (j < NLOG)       v = W_logit[k * NLOG + j];
    else if (j == NLOG) v = W_val[k];
    Wh[idx] = v;
}

// ---------------------------------------------------------------------------
// Head: [logits | value] = X @ W_head (+biases). Block = 64 thr (2 waves),
// wave w covers cols 16w..16w+15; epilogue writes straight into d_out.
// ---------------------------------------------------------------------------
__global__ void k_head_wmma(const float* __restrict__ X,   // [B, NN, 1024]
                            const float* __restrict__ Wh,  // [1024, 32]
                            const float* __restrict__ b_logit,
                            const float* __restrict__ b_val,
                            float* __restrict__ out) {
    const int lane  = threadIdx.x & 31;
    const int wave  = threadIdx.x >> 5;            // 0..1
    const int n0    = wave * 16;
    const int m0    = blockIdx.x * 16;
    const int b     = blockIdx.y;
    const int mlane = lane & 15;
    const int hi    = lane >> 4;

    const float* Xb   = X + (size_t)b * NN * FOUT;
    const float* arow = Xb + (size_t)(m0 + mlane) * FOUT;
    const float* wcol = Wh + n0 + mlane;

    v8f acc = {};
#pragma unroll 8
    for (int k = 0; k < FOUT; k += 4) {
        const int ka = k + 2 * hi;
        v2f a, bb;
        a.x  = arow[ka];
        a.y  = arow[ka + 1];
        bb.x = wcol[(size_t)ka * 32];
        bb.y = wcol[(size_t)(ka + 1) * 32];
        acc = __builtin_amdgcn_wmma_f32_16x16x4_f32(
            false, a, false, bb, (short)0, acc, false, false);
    }

    float* logits = out;                                  // [B, NN, 18]
    float* values = out + (size_t)NB * NN * NLOG;         // [B, NN]
    const int col = n0 + mlane;
#pragma unroll
    for (int r = 0; r < 8; ++r) {
        const int row = m0 + r + 8 * hi;                  // node index
        const float v = acc[r];
        if (col < NLOG) {
            logits[((size_t)b * NN + row) * NLOG + col] = v + b_logit[col];
        } else if (col == NLOG) {
            values[(size_t)b * NN + row] = v + b_val[0];
        }
    }
}

// ---------------------------------------------------------------------------
// Host orchestration
// ---------------------------------------------------------------------------
extern "C" void kernel_launch(void* const* d_in, const int* in_sizes, int n_in,
                              void* d_out, int out_size, void* d_ws, size_t ws_size,
                              hipStream_t stream) {
    (void)in_sizes; (void)n_in; (void)out_size; (void)ws_size;

    const float*     nodes   = (const float*)d_in[0];     // [B, NN, 128]
    const long long* edges   = (const long long*)d_in[1]; // [B, 2, NE] int64
    const float*     W0      = (const float*)d_in[2];     // [128, 256]
    const float*     b0      = (const float*)d_in[3];
    const float*     W1      = (const float*)d_in[4];     // [256, 1024]
    const float*     b1      = (const float*)d_in[5];
    const float*     W_logit = (const float*)d_in[6];     // [1024, 18]
    const float*     b_logit = (const float*)d_in[7];
    const float*     W_val   = (const float*)d_in[8];     // [1024, 1]
    const float*     b_val   = (const float*)d_in[9];
    float* out = (float*)d_out;

    // Workspace layout (floats)
    float* ws   = (float*)d_ws;
    float* deg  = ws;                         ws += (size_t)NB * NN;
    float* dinv = ws;                         ws += (size_t)NB * NN;
    float* Wh   = ws;                         ws += (size_t)FOUT * 32;
    float* h0   = ws;                         ws += (size_t)NB * NN * FH;
    float* agg1 = ws;                         ws += (size_t)NB * NN * FH;
    float* h1   = ws;                         ws += (size_t)NB * NN * FOUT;
    float* agg2 = ws;                         ws += (size_t)NB * NN * FOUT;

    // Head weight packing + degree normalization
    k_prep_head<<<(FOUT * 32 + 255) / 256, 256, 0, stream>>>(W_logit, W_val, Wh);
    k_deg_init<<<(NB * NN + 255) / 256, 256, 0, stream>>>(deg, NB * NN);
    k_deg_accum<<<(NB * NE + 255) / 256, 256, 0, stream>>>(edges, deg);
    k_dinv<<<(NB * NN + 255) / 256, 256, 0, stream>>>(deg, dinv, NB * NN);

    // Layer 1: h0 = X @ W0 ; agg1 = norm-aggregate(h0) ; relu(agg1 + b0)
    gemm_wmma_f32<FOBS, FH><<<dim3(FH / 64, NN / 16, NB), 128, 0, stream>>>(
        nodes, W0, h0, NN);
    {
        size_t tot = (size_t)NB * NN * FH;
        k_self_init<FH><<<(unsigned)((tot + 255) / 256), 256, 0, stream>>>(h0, dinv, agg1);
        k_scatter<FH, 64><<<dim3(NE, NB), 64, 0, stream>>>(edges, h0, dinv, agg1);
        k_bias_relu<FH><<<(unsigned)((tot + 255) / 256), 256, 0, stream>>>(agg1, b0);
    }

    // Layer 2: h1 = agg1 @ W1 ; agg2 = norm-aggregate(h1) ; relu(agg2 + b1)
    gemm_wmma_f32<FH, FOUT><<<dim3(FOUT / 64, NN / 16, NB), 128, 0, stream>>>(
        agg1, W1, h1, NN);
    {
        size_t tot = (size_t)NB * NN * FOUT;
        k_self_init<FOUT><<<(unsigned)((tot + 255) / 256), 256, 0, stream>>>(h1, dinv, agg2);
        k_scatter<FOUT, 256><<<dim3(NE, NB), 256, 0, stream>>>(edges, h1, dinv, agg2);
        k_bias_relu<FOUT><<<(unsigned)((tot + 255) / 256), 256, 0, stream>>>(agg2, b1);
    }

    // Heads: logits + values directly into d_out
    k_head_wmma<<<dim3(NN / 16, NB), 64, 0, stream>>>(agg2, Wh, b_logit, b_val, out);
}